// BigramHash_40054865002781
// MI455X (gfx1250) — compile-verified
//
#include <hip/hip_runtime.h>

typedef __attribute__((ext_vector_type(2))) float v2f;
typedef __attribute__((ext_vector_type(8))) float v8f;

#define HASH_MULT   92821u
#define NUM_BUCKETS 2000003u
#define HASH_DIM    64
#define MODEL_DIM   1024
#define SEQ_LEN     8192

// One block (256 threads = 8 wave32) owns a 16-token M-tile and the full
// N=1024 output width. Wave w computes N-tiles [w*8, w*8+8) of 16 columns.
// D = A(16x64, gathered fp32 embeddings) x B(64x16 slices of proj_w^T) via
// 16 chained V_WMMA_F32_16X16X4_F32 per output tile.
__global__ __launch_bounds__(256)
void bigram_hash_proj_wmma(const int* __restrict__ ids,
                           const float* __restrict__ table,
                           const float* __restrict__ proj,
                           float* __restrict__ out)
{
    const int tid     = threadIdx.x;
    const int lane    = tid & 31;
    const int wave    = tid >> 5;
    const int lane16  = lane & 15;
    const int halfsel = lane >> 4;          // 0: lanes 0-15, 1: lanes 16-31
    const int m0      = blockIdx.x * 16;    // 16 tokens per block

    // ---- bigram hash for this lane's token row (lanes 16-31 mirror 0-15) ----
    const int row = m0 + lane16;
    const unsigned long long cur  = (unsigned long long)(unsigned)ids[row];
    const unsigned long long prev = ((row & (SEQ_LEN - 1)) == 0)
        ? 0ull
        : (unsigned long long)(unsigned)ids[row - 1];
    const unsigned long long h =
        (prev * (unsigned long long)HASH_MULT + cur) % (unsigned long long)NUM_BUCKETS;

    // ---- load all 16 A fragments once (16x4 fp32 per WMMA step) ----
    // A layout: lanes 0-15 hold K = {4k,4k+1}, lanes 16-31 hold K = {4k+2,4k+3}
    const float* ap = table + h * (unsigned long long)HASH_DIM + (halfsel ? 2 : 0);
    v2f a[16];
#pragma unroll
    for (int kk = 0; kk < 16; ++kk)
        a[kk] = __builtin_nontemporal_load((const v2f*)(ap + 4 * kk));

    // ---- 8 N-tiles per wave ----
#pragma unroll
    for (int t = 0; t < 8; ++t) {
        const int n0   = (wave * 8 + t) * 16;
        const int ncol = n0 + lane16;

        // B layout: lane holds column ncol; VGPR0/1 = rows k0,k0+1 of B = proj_w[ncol][k0..k0+1]
        const float* bp = proj + (size_t)ncol * HASH_DIM + (halfsel ? 2 : 0);
        v2f b[16];
#pragma unroll
        for (int kk = 0; kk < 16; ++kk)
            b[kk] = *(const v2f*)(bp + 4 * kk);

        v8f c = {};
#pragma unroll
        for (int kk = 0; kk < 16; ++kk)
            c = __builtin_amdgcn_wmma_f32_16x16x4_f32(
                    /*neg_a=*/false, a[kk],
                    /*neg_b=*/false, b[kk],
                    /*c_mod=*/(short)0, c,
                    /*reuse_a=*/false, /*reuse_b=*/false);

        // C/D layout: VGPR v -> output row (v + halfsel*8), column ncol.
        // Output is a 134 MB stream -> non-temporal stores, keep L2 for proj/table.
        float* op = out + (size_t)(m0 + halfsel * 8) * MODEL_DIM + ncol;
#pragma unroll
        for (int v = 0; v < 8; ++v)
            __builtin_nontemporal_store(c[v], op + (size_t)v * MODEL_DIM);
    }
}

extern "C" void kernel_launch(void* const* d_in, const int* in_sizes, int n_in,
                              void* d_out, int out_size, void* d_ws, size_t ws_size,
                              hipStream_t stream)
{
    const int*   ids   = (const int*)d_in[0];    // input_ids [4, 8192]
    const float* table = (const float*)d_in[1];  // [2000003, 64]
    const float* proj  = (const float*)d_in[2];  // [1024, 64]
    float*       out   = (float*)d_out;          // [4, 8192, 1024]

    const int total_rows = in_sizes[0];          // 32768
    const int mblocks    = total_rows / 16;      // 2048

    bigram_hash_proj_wmma<<<dim3(mblocks), dim3(256), 0, stream>>>(ids, table, proj, out);
}